// FunctionalGaussianProcess_44530220925745
// MI455X (gfx1250) — compile-verified
//
#include <hip/hip_runtime.h>
#include <hip/hip_bf16.h>

// ---------------------------------------------------------------------------
// Functional GP posterior, CDNA5 (gfx1250, wave32).
//
// Heavy stage: C[i,j] = sum_{p,q} A[i,p] * exp(-0.5*||Z_ip - Z_jq||^2) * A[j,q]
// mapped onto V_WMMA_F32_16X16X4_F32 for the Gram tiles (Z_i 32x8 @ Z_j^T 8x32)
// plus v_exp_f32 (TRANS, co-issues with XDL WMMA) for the kernel evaluation.
// Everything is L2-resident (Z=256KB, C=256KB) -> compute-bound; exploit
// symmetry (C = C^T) to halve work, stage the i-side operands in LDS and
// reuse them across 8 waves (one j per wave) per block.
// ---------------------------------------------------------------------------

typedef __attribute__((ext_vector_type(2))) float v2f;
typedef __attribute__((ext_vector_type(8))) float v8f;

#define N_OBS  192
#define M_TEST 64
#define NF     256          // N_OBS + M_TEST
#define KP     32           // support points per function
#define DD     8            // dims
#define NK     (NF * KP)    // 8192 total support points
#define NRHS   65           // 64 Kox columns + ones vector
#define WAVES_PER_BLOCK 8

// workspace float offsets
#define WS_Z    0                       // NK*DD       = 65536
#define WS_S    (WS_Z + NK * DD)        // NK          = 8192
#define WS_A    (WS_S + NK)             // NK          = 8192
#define WS_C    (WS_A + NK)             // NF*NF       = 65536
#define WS_KF   (WS_C + NF * NF)        // NF*NF       = 65536
#define WS_M    (WS_KF + NF * NF)       // N_OBS*N_OBS = 36864
#define WS_B    (WS_M + N_OBS * N_OBS)  // N_OBS*NRHS  = 12480
#define WS_AUX  (WS_B + N_OBS * NRHS)   // 65 (Kbx[64], ssum)
// total ~262.5K floats ~ 1.05 MB

// ---------------------------------------------------------------------------
// Kernel 1: Z = X / rho_base (concat obs+test), S = row sq-norms, A concat.
// ---------------------------------------------------------------------------
__global__ void fgp_prep(const float* __restrict__ X_obs,
                         const float* __restrict__ A_obs,
                         const float* __restrict__ X_test,
                         const float* __restrict__ A_test,
                         const float* __restrict__ rho_base,
                         float* __restrict__ ws) {
    int p = blockIdx.x * blockDim.x + threadIdx.x;   // support point 0..NK-1
    if (p >= NK) return;
    int f = p / KP;
    const float* xs;
    float av;
    if (f < N_OBS) {
        xs = X_obs + (size_t)p * DD;
        av = A_obs[p];
    } else {
        xs = X_test + (size_t)(p - N_OBS * KP) * DD;
        av = A_test[p - N_OBS * KP];
    }
    float s = 0.f;
#pragma unroll
    for (int d = 0; d < DD; ++d) {
        float z = xs[d] / rho_base[d];
        ws[WS_Z + (size_t)p * DD + d] = z;
        s += z * z;
    }
    ws[WS_S + p] = s;
    ws[WS_A + p] = av;
}

// ---------------------------------------------------------------------------
// Kernel 2: block = function i (i-side operands staged in LDS once),
// wave w handles j = blockIdx.y*8 + w; symmetry: only j >= i computed,
// the scalar result is mirrored into C[i,j] and C[j,i].
//
// Per pair: G = Z_i (32x8) @ Z_j^T (8x32) via 4 tiles x 2 chained
// WMMA_F32_16X16X4_F32, then d2 = max(si+sj-2G,0),
// partial += a_i*a_j*exp(-0.5*d2), wave32 xor-reduce.
//
// Fragment layouts per CDNA5 ISA 7.12.2 (wave32):
//   A 16x4 f32 : lane l -> M=l&15 ; VGPR0/1 hold K = 2*(l>>4), 2*(l>>4)+1
//   B 4x16 f32 : lane l -> N=l&15 ; VGPR0/1 hold K = 2*(l>>4), 2*(l>>4)+1
//   C/D 16x16  : VGPR r -> M = r + 8*(l>>4), N = l&15
// ---------------------------------------------------------------------------
__global__ __launch_bounds__(32 * WAVES_PER_BLOCK)
void fgp_pairC(float* __restrict__ ws) {
    __shared__ float sZi[KP * DD];   // 32x8 row-major
    __shared__ float sSi[KP];
    __shared__ float sAi[KP];

    const int i    = blockIdx.x;
    const int tid  = threadIdx.x;
    const int wave = tid >> 5;
    const int lane = tid & 31;
    const int j    = blockIdx.y * WAVES_PER_BLOCK + wave;

    const float* Z = ws + WS_Z;
    const float* S = ws + WS_S;
    const float* A = ws + WS_A;

    // Stage function-i operands (shared by all 8 waves of this block).
    if (tid < KP * DD) sZi[tid] = Z[(size_t)i * KP * DD + tid];
    if (tid < KP) {
        sSi[tid] = S[i * KP + tid];
        sAi[tid] = A[i * KP + tid];
    }
    __syncthreads();

    if (j >= i) {   // wave-uniform: EXEC stays all-ones for WMMA
        const float* Zj = Z + (size_t)j * KP * DD;
        __builtin_prefetch(Zj, 0, 1);           // global_prefetch_b8

        const int half = lane >> 4;
        const int lm   = lane & 15;
        float partial = 0.f;

#pragma unroll
        for (int ti = 0; ti < 2; ++ti) {
            // A fragments for row tile ti, from LDS
            const float* ra = sZi + (ti * 16 + lm) * DD + half * 2;
            v2f a0, a1;
            a0.x = ra[0]; a0.y = ra[1];         // K-slice 0: dims {0,1}/{2,3}
            a1.x = ra[4]; a1.y = ra[5];         // K-slice 1: dims {4,5}/{6,7}
#pragma unroll
            for (int tj = 0; tj < 2; ++tj) {
                const float* rb = Zj + (tj * 16 + lm) * DD + half * 2;
                v2f b0, b1;
                b0.x = rb[0]; b0.y = rb[1];
                b1.x = rb[4]; b1.y = rb[5];

                v8f acc = {};
                acc = __builtin_amdgcn_wmma_f32_16x16x4_f32(
                    false, a0, false, b0, (short)0, acc, false, false);
                acc = __builtin_amdgcn_wmma_f32_16x16x4_f32(
                    false, a1, false, b1, (short)0, acc, false, false);

                const int nq = tj * 16 + lm;    // q index within function j
                const float sj = S[j * KP + nq];
                const float aj = A[j * KP + nq];
#pragma unroll
                for (int r = 0; r < 8; ++r) {
                    const int mp = ti * 16 + half * 8 + r;  // p within func i
                    float d2 = fmaxf(sSi[mp] + sj - 2.0f * acc[r], 0.f);
                    partial += sAi[mp] * aj * __expf(-0.5f * d2);
                }
            }
        }

        // wave32 reduction
#pragma unroll
        for (int off = 16; off > 0; off >>= 1)
            partial += __shfl_xor(partial, off, 32);
        if (lane == 0) {
            ws[WS_C + (size_t)i * NF + j] = partial;
            ws[WS_C + (size_t)j * NF + i] = partial;   // symmetry
        }
    }
}

// ---------------------------------------------------------------------------
// Kernel 3: Kf[i,j] = exp(-0.5 * max(C[i,i]+C[j,j]-2*C[i,j], 0) / rho)
// ---------------------------------------------------------------------------
__global__ void fgp_funcK(float* __restrict__ ws, const float* __restrict__ rho_p) {
    int t = blockIdx.x * blockDim.x + threadIdx.x;
    if (t >= NF * NF) return;
    int i = t / NF, j = t % NF;
    const float* C = ws + WS_C;
    float d2 = fmaxf(C[i * NF + i] + C[j * NF + j] - 2.0f * C[i * NF + j], 0.f);
    ws[WS_KF + t] = __expf(-0.5f * d2 / rho_p[0]);
}

// ---------------------------------------------------------------------------
// Kernel 4: single workgroup solver.  Koo = nu*(Kf_oo + g I)  (SPD).
// Cholesky (lower), solve for 65 RHS ([Kox | 1]), then mean / cov -> d_out.
// mean = b + gain@(ys-b);  cov = Kxx - gain@Kox + (1-Kbx)^T(1-Kbx)/(1^T Koo^-1 1)
// gain[t][i] = Xsol[i][t]  (Xsol = Koo^{-1} Kox).
// ---------------------------------------------------------------------------
__global__ __launch_bounds__(256)
void fgp_solve(float* __restrict__ ws,
               const float* __restrict__ ys,
               const float* __restrict__ g_p,
               const float* __restrict__ nu_p,
               const float* __restrict__ b_p,
               float* __restrict__ out) {
    const int tid = threadIdx.x;
    const int nt  = blockDim.x;
    const int n   = N_OBS;
    float* M   = ws + WS_M;
    float* B   = ws + WS_B;
    float* aux = ws + WS_AUX;
    const float* Kf = ws + WS_KF;
    const float nu = nu_p[0];
    const float g  = g_p[0];
    const float bb = b_p[0];

    // Build Koo and RHS [nu*Kox | ones]
    for (int t = tid; t < n * n; t += nt) {
        int i = t / n, j = t % n;
        M[t] = nu * (Kf[i * NF + j] + (i == j ? g : 0.f));
    }
    for (int t = tid; t < n * NRHS; t += nt) {
        int i = t / NRHS, c = t % NRHS;
        B[t] = (c < M_TEST) ? nu * Kf[i * NF + (n + c)] : 1.0f;
    }
    __syncthreads();

    // Right-looking Cholesky, lower triangle in-place.
    for (int kk = 0; kk < n; ++kk) {
        if (tid == 0) M[kk * n + kk] = sqrtf(M[kk * n + kk]);
        __syncthreads();
        float dk = M[kk * n + kk];
        for (int i = kk + 1 + tid; i < n; i += nt) M[i * n + kk] /= dk;
        __syncthreads();
        int rem = n - kk - 1;
        for (int t = tid; t < rem * rem; t += nt) {
            int i = kk + 1 + t / rem;
            int j = kk + 1 + t % rem;
            if (j <= i) M[i * n + j] -= M[i * n + kk] * M[j * n + kk];
        }
        __syncthreads();
    }

    // Forward substitution: L y = B (one thread per RHS column)
    if (tid < NRHS) {
        int c = tid;
        for (int i = 0; i < n; ++i) {
            float acc = B[i * NRHS + c];
            for (int j = 0; j < i; ++j) acc -= M[i * n + j] * B[j * NRHS + c];
            B[i * NRHS + c] = acc / M[i * n + i];
        }
    }
    __syncthreads();
    // Back substitution: L^T x = y
    if (tid < NRHS) {
        int c = tid;
        for (int i = n - 1; i >= 0; --i) {
            float acc = B[i * NRHS + c];
            for (int j = i + 1; j < n; ++j) acc -= M[j * n + i] * B[j * NRHS + c];
            B[i * NRHS + c] = acc / M[i * n + i];
        }
    }
    __syncthreads();

    // Kbx[t] = sum_i gain[t][i] = sum_i Xsol[i][t];  ssum = 1^T Koo^-1 1
    if (tid < M_TEST) {
        float s = 0.f;
        for (int i = 0; i < n; ++i) s += B[i * NRHS + tid];
        aux[tid] = s;
    }
    if (tid == M_TEST) {
        float s = 0.f;
        for (int i = 0; i < n; ++i) s += B[i * NRHS + M_TEST];
        aux[M_TEST] = s;
    }
    __syncthreads();

    // mean
    if (tid < M_TEST) {
        float acc = 0.f;
        for (int i = 0; i < n; ++i) acc += B[i * NRHS + tid] * (ys[i] - bb);
        out[tid] = bb + acc;
    }

    // cov
    const float ssum = aux[M_TEST];
    for (int t = tid; t < M_TEST * M_TEST; t += nt) {
        int t1 = t / M_TEST, t2 = t % M_TEST;
        float acc = nu * Kf[(n + t1) * NF + (n + t2)];
        for (int i = 0; i < n; ++i)
            acc -= B[i * NRHS + t1] * (nu * Kf[i * NF + (n + t2)]);
        acc += (1.f - aux[t1]) * (1.f - aux[t2]) / ssum;
        out[M_TEST + t1 * M_TEST + t2] = acc;
    }
}

// ---------------------------------------------------------------------------
extern "C" void kernel_launch(void* const* d_in, const int* in_sizes, int n_in,
                              void* d_out, int out_size, void* d_ws, size_t ws_size,
                              hipStream_t stream) {
    const float* X_obs    = (const float*)d_in[0];
    const float* A_obs    = (const float*)d_in[1];
    const float* X_test   = (const float*)d_in[2];
    const float* A_test   = (const float*)d_in[3];
    const float* ys       = (const float*)d_in[4];
    const float* rho_base = (const float*)d_in[5];
    const float* rho      = (const float*)d_in[6];
    const float* g        = (const float*)d_in[7];
    const float* nu       = (const float*)d_in[8];
    const float* b        = (const float*)d_in[9];
    float* ws  = (float*)d_ws;
    float* out = (float*)d_out;

    fgp_prep<<<(NK + 255) / 256, 256, 0, stream>>>(X_obs, A_obs, X_test, A_test,
                                                   rho_base, ws);
    fgp_pairC<<<dim3(NF, NF / WAVES_PER_BLOCK), 32 * WAVES_PER_BLOCK, 0, stream>>>(ws);
    fgp_funcK<<<(NF * NF + 255) / 256, 256, 0, stream>>>(ws, rho);
    fgp_solve<<<1, 256, 0, stream>>>(ws, ys, g, nu, b, out);
}